// Model_31172872634678
// MI455X (gfx1250) — compile-verified
//
#include <hip/hip_runtime.h>
#include <math.h>

#define NUM_NODES 100000
#define NUM_RELS  50
#define NUM_BASES 16
#define H_DIM     16
#define OUT_DIM   8
#define NUM_EDGES 3200000
// capacity for relation-sorted edges, each bin padded to a multiple of 16
#define PAD_EDGES 3200752            // NUM_EDGES + 50*15 rounded up to 16
#define PAD_CHUNKS (PAD_EDGES / 16)

typedef __attribute__((ext_vector_type(2))) float v2f;
typedef __attribute__((ext_vector_type(8))) float v8f;

// ---------------------------------------------------------------------------
// Kernel A: basis combination  Wr[r, :] = sum_b C[r,b] * W[b, :]
// GEMM (M=50 pad 64, K=16, N=ncols) via chained V_WMMA_F32_16X16X4_F32.
// ---------------------------------------------------------------------------
__global__ void basis_combine_wmma(const float* __restrict__ C,   // (NUM_RELS, 16)
                                   const float* __restrict__ W,   // (16, ncols)
                                   float* __restrict__ out,       // (NUM_RELS, ncols)
                                   int ncols) {
  const int ntiles = ncols >> 4;
  const int mtile  = blockIdx.x / ntiles;
  const int ntile  = blockIdx.x % ntiles;
  const int lane   = threadIdx.x & 31;
  const int half   = lane >> 4;
  const int l16    = lane & 15;

  const int m  = mtile * 16 + l16;
  const int mc = m < (NUM_RELS - 1) ? m : (NUM_RELS - 1);
  const int n  = ntile * 16 + l16;

  v8f acc = {};
#pragma unroll
  for (int ks = 0; ks < 4; ++ks) {
    const int k0 = ks * 4 + half * 2;
    v2f a, b;
    float a0 = C[mc * NUM_BASES + k0];
    float a1 = C[mc * NUM_BASES + k0 + 1];
    a.x = (m < NUM_RELS) ? a0 : 0.0f;
    a.y = (m < NUM_RELS) ? a1 : 0.0f;
    b.x = W[k0 * ncols + n];
    b.y = W[(k0 + 1) * ncols + n];
    acc = __builtin_amdgcn_wmma_f32_16x16x4_f32(false, a, false, b,
                                                (short)0, acc, false, false);
  }
#pragma unroll
  for (int v = 0; v < 8; ++v) {
    const int mr = mtile * 16 + half * 8 + v;
    if (mr < NUM_RELS) out[mr * ncols + n] = acc[v];
  }
}

// ---------------------------------------------------------------------------
// Counting sort of edges by relation (3 passes).
// ---------------------------------------------------------------------------
__global__ void hist_edges(const int* __restrict__ etyp, int* __restrict__ gh) {
  __shared__ int h[NUM_RELS];
  for (int i = threadIdx.x; i < NUM_RELS; i += blockDim.x) h[i] = 0;
  __syncthreads();
  const int stride = gridDim.x * blockDim.x;
  for (int e = blockIdx.x * blockDim.x + threadIdx.x; e < NUM_EDGES; e += stride)
    atomicAdd(&h[etyp[e]], 1);
  __syncthreads();
  for (int i = threadIdx.x; i < NUM_RELS; i += blockDim.x) atomicAdd(&gh[i], h[i]);
}

__global__ void scan_bins(const int* __restrict__ gh, int* __restrict__ cursor) {
  if (threadIdx.x == 0) {
    int off = 0;
    for (int r = 0; r < NUM_RELS; ++r) {
      cursor[r] = off;
      off += (gh[r] + 15) & ~15;     // pad each bin to a multiple of 16
    }
  }
}

__global__ void scatter_edges(const int* __restrict__ esrc, const int* __restrict__ edst,
                              const int* __restrict__ etyp, const float* __restrict__ enorm,
                              int* __restrict__ cursor,
                              int* __restrict__ ssrc, int* __restrict__ sdst,
                              int* __restrict__ srel, float* __restrict__ snorm) {
  const int stride = gridDim.x * blockDim.x;
  for (int e = blockIdx.x * blockDim.x + threadIdx.x; e < NUM_EDGES; e += stride) {
    const int r = etyp[e];
    const int p = atomicAdd(&cursor[r], 1);
    ssrc[p]  = esrc[e];
    sdst[p]  = edst[e];
    srel[p]  = r;
    snorm[p] = enorm[e];
  }
}

// ---------------------------------------------------------------------------
// Kernel B: input layer scatter (not a matmul: W0[:,src,:] varies per edge).
// 16 lanes per edge (lane = h index).  W0 gathers are L2-resident (102 MB).
// ---------------------------------------------------------------------------
__global__ void layer0_edges(const int* __restrict__ esrc, const int* __restrict__ edst,
                             const int* __restrict__ etyp, const float* __restrict__ enorm,
                             const float* __restrict__ W0, const float* __restrict__ C0,
                             float* __restrict__ h0acc) {
  const long long stride = (long long)gridDim.x * blockDim.x;
  const long long TOT    = (long long)NUM_EDGES * 16;
  for (long long tid = (long long)blockIdx.x * blockDim.x + threadIdx.x;
       tid < TOT; tid += stride) {
    const long long e = tid >> 4;
    const int h = (int)(tid & 15);
    const int src = esrc[e];
    const int dst = edst[e];
    const int rel = etyp[e];
    const float norm = enorm[e];
    const float clane = C0[rel * NUM_BASES + h];
    const float* wbase = W0 + (long long)src * H_DIM + h;
    float acc = 0.0f;
#pragma unroll
    for (int b = 0; b < NUM_BASES; ++b) {
      const float cb = __shfl(clane, b, 16);
      acc = fmaf(cb, wbase[(long long)b * (NUM_NODES * H_DIM)], acc);
    }
    atomicAdd(&h0acc[(long long)dst * H_DIM + h], acc * norm);
  }
}

// ---------------------------------------------------------------------------
// Dense layer on relation-sorted edges via WMMA.
// One wave = one 16-edge chunk (relation-uniform by construction):
//   msg(16e x OUTC) = relu(H[src_e,:]) (16x16) x Wr[rel] (16xOUTC)
// K=16 -> 4 chained V_WMMA_F32_16X16X4_F32.  Each h element gathered once.
// Zero-padded slots (norm=0, src=dst=rel=0) contribute exactly zero.
// ---------------------------------------------------------------------------
template <int OUTC>
__global__ void dense_wmma_edges(const int* __restrict__ ssrc, const int* __restrict__ sdst,
                                 const int* __restrict__ srel, const float* __restrict__ snorm,
                                 const float* __restrict__ hin, const float* __restrict__ Wr,
                                 float* __restrict__ hout) {
  const int lane = threadIdx.x & 31;
  const int half = lane >> 4;
  const int l16  = lane & 15;
  const int wave  = (blockIdx.x * blockDim.x + threadIdx.x) >> 5;
  const int nwave = (gridDim.x * blockDim.x) >> 5;

  for (int ch = wave; ch < PAD_CHUNKS; ch += nwave) {
    const int base = ch * 16;
    const int rel  = srel[base];                   // uniform within chunk
    const int src  = ssrc[base + l16];             // A-matrix row m = l16
    const float* hrow = hin + (long long)src * H_DIM;
    const float* wb   = Wr + rel * (H_DIM * OUTC);

    v8f acc = {};
#pragma unroll
    for (int ks = 0; ks < 4; ++ks) {
      const int k0 = ks * 4 + half * 2;
      v2f a, b;
      a.x = fmaxf(hrow[k0], 0.0f);                 // fused ReLU on gather
      a.y = fmaxf(hrow[k0 + 1], 0.0f);
      if (OUTC == 16) {
        b.x = wb[k0 * OUTC + l16];
        b.y = wb[(k0 + 1) * OUTC + l16];
      } else {                                     // OUTC==8: columns 8..15 are zero
        b.x = (l16 < OUTC) ? wb[k0 * OUTC + l16] : 0.0f;
        b.y = (l16 < OUTC) ? wb[(k0 + 1) * OUTC + l16] : 0.0f;
      }
      acc = __builtin_amdgcn_wmma_f32_16x16x4_f32(false, a, false, b,
                                                  (short)0, acc, false, false);
    }
    // D layout: vgpr v -> edge (base + half*8 + v), column l16
#pragma unroll
    for (int v = 0; v < 8; ++v) {
      const int e  = base + half * 8 + v;
      const float nm = snorm[e];
      const int dst  = sdst[e];
      if (OUTC == 16) {
        atomicAdd(&hout[(long long)dst * 16 + l16], acc[v] * nm);
      } else if (l16 < OUTC) {
        atomicAdd(&hout[(long long)dst * OUTC + l16], acc[v] * nm);
      }
    }
  }
}

// ---------------------------------------------------------------------------
// Row softmax over 8-wide rows.
// ---------------------------------------------------------------------------
__global__ void softmax_rows(const float* __restrict__ acc2, float* __restrict__ out) {
  const int n = blockIdx.x * blockDim.x + threadIdx.x;
  if (n >= NUM_NODES) return;
  float v[OUT_DIM];
  float m = -INFINITY;
#pragma unroll
  for (int i = 0; i < OUT_DIM; ++i) { v[i] = acc2[n * OUT_DIM + i]; m = fmaxf(m, v[i]); }
  float s = 0.0f;
#pragma unroll
  for (int i = 0; i < OUT_DIM; ++i) { v[i] = expf(v[i] - m); s += v[i]; }
  const float inv = 1.0f / s;
#pragma unroll
  for (int i = 0; i < OUT_DIM; ++i) out[n * OUT_DIM + i] = v[i] * inv;
}

// ---------------------------------------------------------------------------
extern "C" void kernel_launch(void* const* d_in, const int* in_sizes, int n_in,
                              void* d_out, int out_size, void* d_ws, size_t ws_size,
                              hipStream_t stream) {
  const int*   esrc  = (const int*)d_in[0];
  const int*   edst  = (const int*)d_in[1];
  const int*   etyp  = (const int*)d_in[2];
  const float* enorm = (const float*)d_in[3];
  const float* W0    = (const float*)d_in[4];
  const float* C0    = (const float*)d_in[5];
  const float* W1    = (const float*)d_in[6];
  const float* C1    = (const float*)d_in[7];
  const float* W2    = (const float*)d_in[8];
  const float* C2    = (const float*)d_in[9];

  // -------- workspace layout (~67 MB) --------
  float* Wr1   = (float*)d_ws;                                 // 50*256
  float* Wr2   = Wr1 + NUM_RELS * H_DIM * H_DIM;               // 50*128
  float* h0    = Wr2 + NUM_RELS * H_DIM * OUT_DIM;             // N*16
  float* h1    = h0 + (size_t)NUM_NODES * H_DIM;               // N*16
  float* a2    = h1 + (size_t)NUM_NODES * H_DIM;               // N*8
  int*   hist  = (int*)(a2 + (size_t)NUM_NODES * OUT_DIM);     // 64
  int*   curs  = hist + 64;                                    // 64
  int*   ssrc  = curs + 64;                                    // PAD_EDGES
  int*   sdst  = ssrc + PAD_EDGES;                             // PAD_EDGES
  int*   srel  = sdst + PAD_EDGES;                             // PAD_EDGES
  float* snorm = (float*)(srel + PAD_EDGES);                   // PAD_EDGES

  // per-call zeroing (graph-capture-safe memset nodes)
  hipMemsetAsync(h0,   0, (size_t)NUM_NODES * H_DIM * sizeof(float), stream);
  hipMemsetAsync(h1,   0, (size_t)NUM_NODES * H_DIM * sizeof(float), stream);
  hipMemsetAsync(a2,   0, (size_t)NUM_NODES * OUT_DIM * sizeof(float), stream);
  hipMemsetAsync(hist, 0, 64 * sizeof(int), stream);
  hipMemsetAsync(ssrc, 0, (size_t)PAD_EDGES * 4 * sizeof(int), stream);  // ssrc..snorm contiguous

  // basis combination GEMMs via WMMA (M padded 50->64)
  basis_combine_wmma<<<4 * (H_DIM * H_DIM / 16), 32, 0, stream>>>(C1, W1, Wr1, H_DIM * H_DIM);
  basis_combine_wmma<<<4 * (H_DIM * OUT_DIM / 16), 32, 0, stream>>>(C2, W2, Wr2, H_DIM * OUT_DIM);

  // counting sort by relation (bins padded to multiples of 16)
  hist_edges  <<<512, 256, 0, stream>>>(etyp, hist);
  scan_bins   <<<1, 32, 0, stream>>>(hist, curs);
  scatter_edges<<<2048, 256, 0, stream>>>(esrc, edst, etyp, enorm, curs,
                                          ssrc, sdst, srel, snorm);

  // layer 0: scalar gather/scatter (L2-resident W0)
  layer0_edges<<<8192, 256, 0, stream>>>(esrc, edst, etyp, enorm, W0, C0, h0);

  // dense layers on sorted edges: one wave per 16-edge chunk, WMMA matmul
  dense_wmma_edges<16><<<4096, 256, 0, stream>>>(ssrc, sdst, srel, snorm, h0, Wr1, h1);
  dense_wmma_edges<8> <<<4096, 256, 0, stream>>>(ssrc, sdst, srel, snorm, h1, Wr2, a2);

  softmax_rows<<<(NUM_NODES + 255) / 256, 256, 0, stream>>>(a2, (float*)d_out);
}